// Net_64871186038856
// MI455X (gfx1250) — compile-verified
//
#include <hip/hip_runtime.h>

// MI455X / gfx1250: fp32 WMMA (V_WMMA_F32_16X16X4_F32) MLP evaluator.
// Compute-bound (~42 GFLOP over ~25 MB of I/O). Each wave owns 32 batch rows
// (two 16-row tiles sharing weight loads; two independent 3-WMMA chains per
// layer interleave for in-wave latency hiding). Transposed formulation:
// D[16x16] = W^T(A) x h^T(B), bias folded into the K=10 slot (A row = b[j],
// B row = ones). Layer-to-layer operand re-layout uses only half-wave
// shuffles (lane-permute HW). A prolog kernel repacks W_h/b_h into the exact
// per-lane A-operand layout in d_ws, pre-scaled by -log2(e) so the per-layer
// sigmoid is just rcp(1 + exp2(d)): the hot loop then needs only one b128 +
// one b64 load per layer with loop-invariant voffsets.

typedef __attribute__((ext_vector_type(2))) float v2f;
typedef __attribute__((ext_vector_type(4))) float v4f;
typedef __attribute__((ext_vector_type(8))) float v8f;

#define NLAYERS 100
#define LOG2E 1.44269504088896340736f
#define WS_FLOATS (NLAYERS * 192)  // 76,800 B of d_ws used

__device__ __forceinline__ float sig_pre(float d) {
  // d is already -log2(e) * preactivation (scale folded into weights)
  return __builtin_amdgcn_rcpf(1.0f + __builtin_amdgcn_exp2f(d));
}
__device__ __forceinline__ float fast_sigmoid(float x) {
  return __builtin_amdgcn_rcpf(1.0f + __builtin_amdgcn_exp2f(x * -LOG2E));
}
__device__ __forceinline__ float xsw16(float v) {
  // swap lane halves (0-15 <-> 16-31); executed by all lanes (no divergence)
  return __shfl_xor(v, 16, 32);
}

#define WMMA4(A, B, C)                                                        \
  __builtin_amdgcn_wmma_f32_16x16x4_f32(false, (A), false, (B), (short)0,     \
                                        (C), false, false)

// ---------------------------------------------------------------------------
// Prolog: repack [100,10,10] W_h + [100,10] b_h into per-lane A-operand
// layout, one 192-float block per layer:
//   floats [  0..127]: lane*4 + {0,1,2,3} = A0.x A0.y A1.x A1.y  (K chunks 0,1)
//   floats [128..191]: lane*2 + {0,1}     = A2.x A2.y            (K chunk 2)
// Slot (c,v) at lane holds W[L][i][j] with i = 4c + v + 2*(lane>=16), j=lane&15;
// i==10 -> bias b[j] (ones row in B), i==11 or j>=10 -> 0. All values scaled
// by -log2(e) so sigmoid needs no multiply.
// ---------------------------------------------------------------------------
__global__ __launch_bounds__(256) void prep_weights(const float* __restrict__ Wh,
                                                    const float* __restrict__ bh,
                                                    float* __restrict__ ws) {
  const int t = blockIdx.x * blockDim.x + threadIdx.x;
  if (t >= WS_FLOATS) return;
  const int L = t / 192;
  const int idx = t - L * 192;
  int lane, c, v;
  if (idx < 128) {
    lane = idx >> 2;
    c = (idx & 3) >> 1;
    v = idx & 1;
  } else {
    const int k = idx - 128;
    lane = k >> 1;
    c = 2;
    v = k & 1;
  }
  const int j = lane & 15;
  const bool lo = lane < 16;
  const int i = 4 * c + v + (lo ? 0 : 2);  // K-row of W^T this slot holds
  float val = 0.f;
  if (j < 10) {
    if (i == 10)     val = bh[L * 10 + j];          // K=10: bias row
    else if (i <= 9) val = Wh[L * 100 + i * 10 + j];
    // i == 11: dead column (B row 11 is zero) -> stays 0
  }
  ws[t] = val * -LOG2E;
}

// ---------------------------------------------------------------------------
// Main kernel
// ---------------------------------------------------------------------------
__global__ __launch_bounds__(256) void mlp100_wmma_f32(
    const float* __restrict__ x,    // [B,2]
    const float* __restrict__ Win,  // [2,10]
    const float* __restrict__ bin,  // [10]
    const float* __restrict__ ws,   // repacked, pre-scaled hidden weights
    const float* __restrict__ Wout, // [10,1]
    const float* __restrict__ bout, // [1]
    float* __restrict__ out,        // [B,1]
    int rows) {
  const int lane = threadIdx.x & 31;
  const int wave = (int)((blockIdx.x * blockDim.x + threadIdx.x) >> 5);
  const int rowbase = wave << 5;  // 32 rows per wave (2 tiles of 16)
  if (rowbase >= rows) return;    // uniform per wave -> EXEC stays all-1s

  const int j = lane & 15;
  const bool lo = lane < 16;
  const int jc = (j < 10) ? j : (j - 10);  // clamped index for dead lanes

  // ---------------- input layer (unscaled; no activation) -----------------
  const float2* x2 = reinterpret_cast<const float2*>(x);
  const float2 xv0 = x2[rowbase + j];
  const float2 xv1 = x2[rowbase + 16 + j];
  v2f A, B0, B1;
  A.x = lo ? Win[jc] : bin[jc];  // K=0: W_in[0][j] | K=2: b_in[j]
  A.y = Win[10 + jc];            // K=1: W_in[1][j] | K=3: dead (B row 3 = 0)
  B0.x = lo ? xv0.x : 1.0f;
  B0.y = lo ? xv0.y : 0.0f;
  B1.x = lo ? xv1.x : 1.0f;
  B1.y = lo ? xv1.y : 0.0f;
  v8f z = {};
  v8f c0 = WMMA4(A, B0, z);
  v8f c1 = WMMA4(A, B1, z);
  // D layout: reg r -> row r in lanes 0-15, row r+8 in lanes 16-31
  float h0 = c0[0], h1 = c0[1], h2 = c0[2], h3 = c0[3];
  float h4 = c0[4], h5 = c0[5], h6 = c0[6], h7 = c0[7];
  float g0 = c1[0], g1 = c1[1], g2 = c1[2], g3 = c1[3];
  float g4 = c1[4], g5 = c1[5], g6 = c1[6], g7 = c1[7];

  const float* Ws = ws;
  const int v4off = lane * 4;  // loop-invariant per-lane element offsets
  const int v2off = 128 + lane * 2;
  for (int L = 0; L < NLAYERS; ++L) {
    // one b128 + one b64 load per layer, scalar base stepped by 192 floats
    const v4f Aq = *reinterpret_cast<const v4f*>(Ws + v4off);
    const v2f A2 = *reinterpret_cast<const v2f*>(Ws + v2off);
    Ws += 192;
    v2f A0, A1;
    A0.x = Aq.x; A0.y = Aq.y;
    A1.x = Aq.z; A1.y = Aq.w;

    // tile 0: rebuild B (h^T) from D registers via half-swaps
    float t2 = xsw16(h2), t3 = xsw16(h3), t6 = xsw16(h6), t7 = xsw16(h7);
    float t0 = xsw16(h0), t1 = xsw16(h1);  // rows 8,9 into low half
    v2f P0, P1, P2;
    P0.x = lo ? h0 : t2;   P0.y = lo ? h1 : t3;   // K = 0,1 | 2,3
    P1.x = lo ? h4 : t6;   P1.y = lo ? h5 : t7;   // K = 4,5 | 6,7
    P2.x = lo ? t0 : 1.0f; P2.y = lo ? t1 : 0.0f; // K = 8,9 | ones, 0

    // tile 1
    float u2 = xsw16(g2), u3 = xsw16(g3), u6 = xsw16(g6), u7 = xsw16(g7);
    float u0 = xsw16(g0), u1 = xsw16(g1);
    v2f Q0, Q1, Q2;
    Q0.x = lo ? g0 : u2;   Q0.y = lo ? g1 : u3;
    Q1.x = lo ? g4 : u6;   Q1.y = lo ? g5 : u7;
    Q2.x = lo ? u0 : 1.0f; Q2.y = lo ? u1 : 0.0f;

    // two independent 3-WMMA chains (interleave to hide WMMA latency)
    v8f d0 = WMMA4(A0, P0, z);
    v8f d1 = WMMA4(A0, Q0, z);
    d0 = WMMA4(A1, P1, d0);
    d1 = WMMA4(A1, Q1, d1);
    d0 = WMMA4(A2, P2, d0);
    d1 = WMMA4(A2, Q2, d1);

    // weights pre-scaled by -log2(e): sigmoid = rcp(1 + exp2(d))
    h0 = sig_pre(d0[0]); h1 = sig_pre(d0[1]);
    h2 = sig_pre(d0[2]); h3 = sig_pre(d0[3]);
    h4 = sig_pre(d0[4]); h5 = sig_pre(d0[5]);
    h6 = sig_pre(d0[6]); h7 = sig_pre(d0[7]);
    g0 = sig_pre(d1[0]); g1 = sig_pre(d1[1]);
    g2 = sig_pre(d1[2]); g3 = sig_pre(d1[3]);
    g4 = sig_pre(d1[4]); g5 = sig_pre(d1[5]);
    g6 = sig_pre(d1[6]); g7 = sig_pre(d1[7]);
  }

  // ---------------- output layer: 10 scalar FMAs + sigmoid per tile -------
  const float w0 = Wout[0], w1 = Wout[1], w2 = Wout[2], w3 = Wout[3];
  const float w4 = Wout[4], w5 = Wout[5], w6 = Wout[6], w7 = Wout[7];
  const float w8o = Wout[8], w9o = Wout[9], b_o = bout[0];

  float r8 = xsw16(h0), r9 = xsw16(h1);  // tile0 rows 8,9 to low half
  float s8 = xsw16(g0), s9 = xsw16(g1);  // tile1
  float o0v = b_o, o1v = b_o;
  o0v = fmaf(w0, h0, o0v);  o1v = fmaf(w0, g0, o1v);
  o0v = fmaf(w1, h1, o0v);  o1v = fmaf(w1, g1, o1v);
  o0v = fmaf(w2, h2, o0v);  o1v = fmaf(w2, g2, o1v);
  o0v = fmaf(w3, h3, o0v);  o1v = fmaf(w3, g3, o1v);
  o0v = fmaf(w4, h4, o0v);  o1v = fmaf(w4, g4, o1v);
  o0v = fmaf(w5, h5, o0v);  o1v = fmaf(w5, g5, o1v);
  o0v = fmaf(w6, h6, o0v);  o1v = fmaf(w6, g6, o1v);
  o0v = fmaf(w7, h7, o0v);  o1v = fmaf(w7, g7, o1v);
  o0v = fmaf(w8o, r8, o0v); o1v = fmaf(w8o, s8, o1v);
  o0v = fmaf(w9o, r9, o0v); o1v = fmaf(w9o, s9, o1v);
  o0v = fast_sigmoid(o0v);
  o1v = fast_sigmoid(o1v);

  // full-wave coalesced store: lanes 0-15 -> tile0, lanes 16-31 -> tile1
  float o1s = xsw16(o1v);
  out[rowbase + lane] = lo ? o0v : o1s;
}

extern "C" void kernel_launch(void* const* d_in, const int* in_sizes, int n_in,
                              void* d_out, int out_size, void* d_ws, size_t ws_size,
                              hipStream_t stream) {
  const float* x    = (const float*)d_in[0];
  const float* Win  = (const float*)d_in[1];
  const float* bin  = (const float*)d_in[2];
  const float* Wh   = (const float*)d_in[3];
  const float* bh   = (const float*)d_in[4];
  const float* Wout = (const float*)d_in[5];
  const float* bout = (const float*)d_in[6];
  float* out = (float*)d_out;
  float* ws  = (float*)d_ws;  // needs WS_FLOATS*4 = 76,800 B of scratch

  // 1) repack + pre-scale hidden-layer weights into d_ws (same stream ->
  //    ordered before the main kernel; re-run every call, graph-capture safe)
  prep_weights<<<(WS_FLOATS + 255) / 256, 256, 0, stream>>>(Wh, bh, ws);

  // 2) main MLP evaluation
  const int rows = in_sizes[0] / 2;        // B
  const int waves = (rows + 31) / 32;      // 32 rows per wave
  const int blocks = (waves + 7) / 8;      // 8 wave32 per 256-thread block
  mlp100_wmma_f32<<<blocks, 256, 0, stream>>>(x, Win, bin, ws, Wout, bout,
                                              out, rows);
}